// SAAtomCenteredDescriptor_15676630631135
// MI455X (gfx1250) — compile-verified
//
#include <hip/hip_runtime.h>
#include <hip/hip_bf16.h>
#include <math.h>

typedef float v2f __attribute__((ext_vector_type(2)));
typedef float v4f __attribute__((ext_vector_type(4)));
typedef float v8f __attribute__((ext_vector_type(8)));

#define PI_F 3.14159265358979323846f
#define CUTOFF_F 5.0f

// ---------------------------------------------------------------- utilities

__device__ __forceinline__ void atomic_max_f32(float* addr, float val) {
  // standard monotonic-bit trick; segmax is initialized to -inf (0xFF800000)
  if (val >= 0.0f) atomicMax((int*)addr, __float_as_int(val));
  else             atomicMin((unsigned int*)addr, (unsigned int)__float_as_int(val));
}

__global__ void fill_f32_kernel(float* __restrict__ p, float v, long long n) {
  long long i = (long long)blockIdx.x * blockDim.x + threadIdx.x;
  if (i < n) p[i] = v;
}

__global__ void fill_u32_kernel(unsigned int* __restrict__ p, unsigned int v, long long n) {
  long long i = (long long)blockIdx.x * blockDim.x + threadIdx.x;
  if (i < n) p[i] = v;
}

// ------------------------------------------------- edge preprocessing kernel
// Per edge: d, unit vector, 16 sinc-RBF values (b0), 25 real-SH values, and
// y_edge = (rbf * emb[Zj]) @ W_rb scattered into y[idx_i, 0, :].
__global__ void edge_pre_kernel(const int* __restrict__ idx,
                                const int* __restrict__ Z,
                                const float* __restrict__ disp,
                                const float* __restrict__ emb,
                                const float* __restrict__ W_rb,
                                float* __restrict__ b0,
                                float* __restrict__ sh,
                                float* __restrict__ y,
                                int E_) {
  int e = blockIdx.x * blockDim.x + threadIdx.x;
  if (e >= E_) return;
  int i = idx[2 * e + 0];
  int j = idx[2 * e + 1];
  float dx = disp[3 * e + 0], dy = disp[3 * e + 1], dz = disp[3 * e + 2];
  float d = sqrtf(dx * dx + dy * dy + dz * dz);
  float dsafe = fmaxf(d, 1e-9f);
  float inv = 1.0f / dsafe;
  float x = dx * inv, yv = dy * inv, z = dz * inv;

  // sinc radial basis with cosine envelope
  float env = 0.5f * (cosf(PI_F * fminf(dsafe, CUTOFF_F) / CUTOFF_F) + 1.0f);
  float r[16];
#pragma unroll
  for (int kk = 0; kk < 16; ++kk) {
    float t = (float)(kk + 1) * dsafe / CUTOFF_F;
    float s = (t == 0.0f) ? 1.0f : sinf(PI_F * t) / (PI_F * t);
    r[kk] = s * env;
  }
  // vectorized b0 store (4x b128)
#pragma unroll
  for (int q4 = 0; q4 < 4; ++q4) {
    v4f rv = {r[4 * q4 + 0], r[4 * q4 + 1], r[4 * q4 + 2], r[4 * q4 + 3]};
    *(v4f*)(b0 + (size_t)e * 16 + 4 * q4) = rv;
  }

  // real spherical harmonics l=0..4 (25 values)
  float x2 = x * x, y2 = yv * yv, z2 = z * z;
  float shv[25];
  shv[0]  = 0.28209479177387814f;
  shv[1]  = 0.4886025119029199f * yv;
  shv[2]  = 0.4886025119029199f * z;
  shv[3]  = 0.4886025119029199f * x;
  shv[4]  = 1.0925484305920792f * x * yv;
  shv[5]  = 1.0925484305920792f * yv * z;
  shv[6]  = 0.31539156525252005f * (3.0f * z2 - 1.0f);
  shv[7]  = 1.0925484305920792f * x * z;
  shv[8]  = 0.5462742152960396f * (x2 - y2);
  shv[9]  = 0.5900435899266435f * yv * (3.0f * x2 - y2);
  shv[10] = 2.890611442640554f * x * yv * z;
  shv[11] = 0.4570457994644658f * yv * (5.0f * z2 - 1.0f);
  shv[12] = 0.3731763325901154f * z * (5.0f * z2 - 3.0f);
  shv[13] = 0.4570457994644658f * x * (5.0f * z2 - 1.0f);
  shv[14] = 1.445305721320277f * z * (x2 - y2);
  shv[15] = 0.5900435899266435f * x * (x2 - 3.0f * y2);
  shv[16] = 2.5033429417967046f * x * yv * (x2 - y2);
  shv[17] = 1.7701307697799304f * yv * z * (3.0f * x2 - y2);
  shv[18] = 0.9461746957575601f * x * yv * (7.0f * z2 - 1.0f);
  shv[19] = 0.6690465435572892f * yv * z * (7.0f * z2 - 3.0f);
  shv[20] = 0.10578554691520431f * (35.0f * z2 * z2 - 30.0f * z2 + 3.0f);
  shv[21] = 0.6690465435572892f * x * z * (7.0f * z2 - 3.0f);
  shv[22] = 0.47308734787878004f * (x2 - y2) * (7.0f * z2 - 1.0f);
  shv[23] = 1.7701307697799304f * x * z * (x2 - 3.0f * y2);
  shv[24] = 0.6258357354491761f * (x2 * x2 - 6.0f * x2 * y2 + y2 * y2);
#pragma unroll
  for (int q4 = 0; q4 < 6; ++q4) {
    v4f sv = {shv[4 * q4 + 0], shv[4 * q4 + 1], shv[4 * q4 + 2], shv[4 * q4 + 3]};
    *(v4f*)(sh + (size_t)e * 25 + 4 * q4) = sv;
  }
  sh[(size_t)e * 25 + 24] = shv[24];

  // y_edge = (rbf * emb[Zj]) @ W_rb  -> scatter into y[i, 0, :]
  int zj = Z[j];
  float t16[16];
#pragma unroll
  for (int f = 0; f < 16; ++f) t16[f] = r[f] * emb[zj * 16 + f];
#pragma unroll
  for (int fo = 0; fo < 16; ++fo) {
    float acc = 0.0f;
#pragma unroll
    for (int f = 0; f < 16; ++f) acc += t16[f] * W_rb[f * 16 + fo];
    atomicAdd(&y[(size_t)i * 400 + fo], acc);
  }
}

// ---------------------------------------------- WMMA f32 projection kernels
// Computes Q/K/V = Y @ W{q,k,v} over M = ntiles*16 rows of 16 features.
// One wave handles a 16x16 tile with 4x V_WMMA_F32_16X16X4_F32 per output.
// A layout (16x4 f32): lane-half hi, VGPR j -> K = 4*kk + 2*hi + j (contig f2).
// B layout (4x16 f32): lane n = lane&15, VGPR j -> W[4*kk + 2*hi + j][n].
// C/D layout: VGPR i -> row (i + 8*hi), col n.
__global__ void qkv_proj_wmma(const float* __restrict__ Y,
                              const float* __restrict__ Wq,
                              const float* __restrict__ Wk,
                              const float* __restrict__ Wv,
                              float* __restrict__ Q,
                              float* __restrict__ Kout,
                              float* __restrict__ V,
                              int ntiles) {
  const int lane = threadIdx.x & 31;
  const int wave = (int)(((long long)blockIdx.x * blockDim.x + threadIdx.x) >> 5);
  const int nwaves = (int)(((long long)gridDim.x * blockDim.x) >> 5);
  const int lm = lane & 15;  // A-row within tile / B,D column
  const int hi = lane >> 4;  // lane-half

  v2f bq[4], bk[4], bv[4];
#pragma unroll
  for (int kk = 0; kk < 4; ++kk) {
    int kr = 4 * kk + 2 * hi;
    bq[kk][0] = Wq[(kr + 0) * 16 + lm];
    bq[kk][1] = Wq[(kr + 1) * 16 + lm];
    bk[kk][0] = Wk[(kr + 0) * 16 + lm];
    bk[kk][1] = Wk[(kr + 1) * 16 + lm];
    bv[kk][0] = Wv[(kr + 0) * 16 + lm];
    bv[kk][1] = Wv[(kr + 1) * 16 + lm];
  }
  for (int t = wave; t < ntiles; t += nwaves) {
    const float* ap = Y + (size_t)t * 256 + lm * 16;  // this lane's row
    v8f cq = {};
    v8f ck = {};
    v8f cv = {};
#pragma unroll
    for (int kk = 0; kk < 4; ++kk) {
      v2f a = *(const v2f*)(ap + 4 * kk + 2 * hi);
      cq = __builtin_amdgcn_wmma_f32_16x16x4_f32(false, a, false, bq[kk], (short)0, cq, false, false);
      ck = __builtin_amdgcn_wmma_f32_16x16x4_f32(false, a, false, bk[kk], (short)0, ck, false, false);
      cv = __builtin_amdgcn_wmma_f32_16x16x4_f32(false, a, false, bv[kk], (short)0, cv, false, false);
    }
    float* qp = Q + (size_t)t * 256;
    float* kp = Kout + (size_t)t * 256;
    float* vp = V + (size_t)t * 256;
#pragma unroll
    for (int i2 = 0; i2 < 8; ++i2) {
      int row = i2 + 8 * hi;
      qp[row * 16 + lm] = cq[i2];
      kp[row * 16 + lm] = ck[i2];
      vp[row * 16 + lm] = cv[i2];
    }
  }
}

// y_tile += agg_tile @ Wo : load C from Y, 4 WMMAs, store D back.
__global__ void wo_accum_wmma(const float* __restrict__ Agg,
                              const float* __restrict__ Wo,
                              float* __restrict__ Y,
                              int ntiles) {
  const int lane = threadIdx.x & 31;
  const int wave = (int)(((long long)blockIdx.x * blockDim.x + threadIdx.x) >> 5);
  const int nwaves = (int)(((long long)gridDim.x * blockDim.x) >> 5);
  const int lm = lane & 15;
  const int hi = lane >> 4;

  v2f bo[4];
#pragma unroll
  for (int kk = 0; kk < 4; ++kk) {
    int kr = 4 * kk + 2 * hi;
    bo[kk][0] = Wo[(kr + 0) * 16 + lm];
    bo[kk][1] = Wo[(kr + 1) * 16 + lm];
  }
  for (int t = wave; t < ntiles; t += nwaves) {
    float* yp = Y + (size_t)t * 256;
    const float* ap = Agg + (size_t)t * 256 + lm * 16;
    v8f c;
#pragma unroll
    for (int i2 = 0; i2 < 8; ++i2) c[i2] = yp[(i2 + 8 * hi) * 16 + lm];
#pragma unroll
    for (int kk = 0; kk < 4; ++kk) {
      v2f a = *(const v2f*)(ap + 4 * kk + 2 * hi);
      c = __builtin_amdgcn_wmma_f32_16x16x4_f32(false, a, false, bo[kk], (short)0, c, false, false);
    }
#pragma unroll
    for (int i2 = 0; i2 < 8; ++i2) yp[(i2 + 8 * hi) * 16 + lm] = c[i2];
  }
}

// scalar tail for rows not covered by full 16-row WMMA tiles
__global__ void proj_tail_kernel(const float* __restrict__ Yin,
                                 const float* __restrict__ W,
                                 float* __restrict__ Out,
                                 int row0, int nrows, int accumulate) {
  int t = blockIdx.x * blockDim.x + threadIdx.x;
  if (t >= nrows * 16) return;
  int r = row0 + (t >> 4), c = t & 15;
  float acc = 0.0f;
#pragma unroll
  for (int f = 0; f < 16; ++f) acc += Yin[(size_t)r * 16 + f] * W[f * 16 + c];
  if (accumulate) Out[(size_t)r * 16 + c] += acc;
  else            Out[(size_t)r * 16 + c]  = acc;
}

// ----------------------------------------------------- attention (3 passes)
// Gather-bound against L2: use b128 loads for the q.k reduction (4x fewer
// vmem instructions than scalar loads).
__global__ void attn_logits_kernel(const int* __restrict__ idx,
                                   const float* __restrict__ b0,
                                   const float* __restrict__ Q,
                                   const float* __restrict__ Kin,
                                   const float* __restrict__ Wb_s,
                                   const float* __restrict__ Wg_s,
                                   const float* __restrict__ Wg2_s,
                                   float* __restrict__ logits,
                                   float* __restrict__ g,
                                   float* __restrict__ g2,
                                   float* __restrict__ segmax,
                                   float scale, int E_) {
  int e = blockIdx.x * blockDim.x + threadIdx.x;
  if (e >= E_) return;
  int i = idx[2 * e + 0];
  int j = idx[2 * e + 1];

  // b0-driven per-head bias / gates (b128 b0 loads)
  v4f b0v[4];
#pragma unroll
  for (int q4 = 0; q4 < 4; ++q4) b0v[q4] = *(const v4f*)(b0 + (size_t)e * 16 + 4 * q4);
  float bias[4];
#pragma unroll
  for (int h = 0; h < 4; ++h) {
    float bb = 0.0f, gg = 0.0f, gg2 = 0.0f;
#pragma unroll
    for (int f = 0; f < 16; ++f) {
      float bv = b0v[f >> 2][f & 3];
      bb  += bv * Wb_s[f * 4 + h];
      gg  += bv * Wg_s[f * 4 + h];
      gg2 += bv * Wg2_s[f * 4 + h];
    }
    bias[h] = bb;
    g[(size_t)e * 4 + h] = gg;
    g2[(size_t)e * 4 + h] = gg2;
  }

  // per-head q.k over L*FH = 100 elements, vectorized: head h owns float4
  // slot h of each 16-float row.
  float acc0 = 0.0f, acc1 = 0.0f, acc2 = 0.0f, acc3 = 0.0f;
  const v4f* qi = (const v4f*)(Q + (size_t)i * 400);
  const v4f* kj = (const v4f*)(Kin + (size_t)j * 400);
#pragma unroll 5
  for (int l = 0; l < 25; ++l) {
    v4f q0 = qi[l * 4 + 0], k0 = kj[l * 4 + 0];
    v4f q1 = qi[l * 4 + 1], k1 = kj[l * 4 + 1];
    v4f q2 = qi[l * 4 + 2], k2 = kj[l * 4 + 2];
    v4f q3 = qi[l * 4 + 3], k3 = kj[l * 4 + 3];
    acc0 += q0[0] * k0[0] + q0[1] * k0[1] + q0[2] * k0[2] + q0[3] * k0[3];
    acc1 += q1[0] * k1[0] + q1[1] * k1[1] + q1[2] * k1[2] + q1[3] * k1[3];
    acc2 += q2[0] * k2[0] + q2[1] * k2[1] + q2[2] * k2[2] + q2[3] * k2[3];
    acc3 += q3[0] * k3[0] + q3[1] * k3[1] + q3[2] * k3[2] + q3[3] * k3[3];
  }
  float accs[4] = {acc0, acc1, acc2, acc3};
#pragma unroll
  for (int h = 0; h < 4; ++h) {
    float lg = accs[h] * scale + bias[h];
    logits[(size_t)e * 4 + h] = lg;
    atomic_max_f32(&segmax[(size_t)i * 4 + h], lg);
  }
}

__global__ void attn_exp_kernel(const int* __restrict__ idx,
                                float* __restrict__ logits,
                                const float* __restrict__ segmax,
                                float* __restrict__ segsum, int E_) {
  int t = blockIdx.x * blockDim.x + threadIdx.x;
  if (t >= E_ * 4) return;
  int e = t >> 2, h = t & 3;
  int i = idx[2 * e];
  float ex = expf(logits[t] - segmax[(size_t)i * 4 + h]);
  logits[t] = ex;  // reuse buffer as exp values
  atomicAdd(&segsum[(size_t)i * 4 + h], ex);
}

// ------------------------------------------------- message + scatter kernel
__global__ void message_kernel(const int* __restrict__ idx,
                               const float* __restrict__ sh,
                               const float* __restrict__ g,
                               const float* __restrict__ g2,
                               const float* __restrict__ ex,
                               const float* __restrict__ segsum,
                               const float* __restrict__ V,
                               float* __restrict__ agg,
                               long long total) {
  long long t = (long long)blockIdx.x * blockDim.x + threadIdx.x;
  if (t >= total) return;
  int f = (int)(t & 15);
  int l = (int)((t >> 4) % 25);
  int e = (int)(t / 400);
  int h = f >> 2;
  int i = idx[2 * e + 0];
  int j = idx[2 * e + 1];
  float attn = ex[(size_t)e * 4 + h] / (segsum[(size_t)i * 4 + h] + 1e-9f);
  float vj0 = V[(size_t)j * 400 + f];
  float vjl = V[(size_t)j * 400 + l * 16 + f];
  float mv = sh[(size_t)e * 25 + l] * (g[(size_t)e * 4 + h] * vj0) +
             g2[(size_t)e * 4 + h] * vjl;
  atomicAdd(&agg[(size_t)i * 400 + l * 16 + f], mv * attn);
}

// ----------------------------------------------------------- final embed add
__global__ void final_embed_kernel(const int* __restrict__ Z,
                                   const float* __restrict__ emb,
                                   const float* __restrict__ W_embed,
                                   const float* __restrict__ b_embed,
                                   float* __restrict__ Y, int n_) {
  int t = blockIdx.x * blockDim.x + threadIdx.x;
  if (t >= n_ * 16) return;
  int fo = t & 15, i = t >> 4;
  int z = Z[i];
  float acc = b_embed[fo];
#pragma unroll
  for (int f = 0; f < 16; ++f) acc += emb[z * 16 + f] * W_embed[f * 16 + fo];
  Y[(size_t)i * 400 + fo] += acc;
}

// ------------------------------------------------------------------- launch
extern "C" void kernel_launch(void* const* d_in, const int* in_sizes, int n_in,
                              void* d_out, int out_size, void* d_ws, size_t ws_size,
                              hipStream_t stream) {
  const int*   Z       = (const int*)d_in[0];
  const int*   idx     = (const int*)d_in[1];
  const float* disp    = (const float*)d_in[2];
  const float* emb     = (const float*)d_in[3];
  const float* W_embed = (const float*)d_in[4];
  const float* b_embed = (const float*)d_in[5];
  const float* W_rb    = (const float*)d_in[6];
  const float* Wq      = (const float*)d_in[7];
  const float* Wk      = (const float*)d_in[8];
  const float* Wv      = (const float*)d_in[9];
  const float* Wo      = (const float*)d_in[10];
  const float* Wb      = (const float*)d_in[11];
  const float* Wg      = (const float*)d_in[12];
  const float* Wg2     = (const float*)d_in[13];

  const int n = in_sizes[0];
  const int E = in_sizes[1] / 2;
  float* y = (float*)d_out;  // [n, 25, 16]

  // workspace layout (floats)
  float* w = (float*)d_ws;
  size_t off = 0;
  const size_t nlf = (size_t)n * 25 * 16;
  float* Q   = w + off; off += nlf;
  float* K   = w + off; off += nlf;
  float* V   = w + off; off += nlf;
  float* AGG = w + off; off += nlf;
  float* B0  = w + off; off += (size_t)E * 16;
  float* SH  = w + off; off += (size_t)E * 25 + 3;  // +3 pad for v4f stores
  float* LG  = w + off; off += (size_t)E * 4;
  float* G   = w + off; off += (size_t)E * 4;
  float* G2  = w + off; off += (size_t)E * 4;
  float* SMX = w + off; off += (size_t)n * 4;
  float* SSM = w + off; off += (size_t)n * 4;
  if (ws_size < off * sizeof(float)) return;  // insufficient scratch

  const float scale = 0.1f;  // 1/sqrt(L*FH) = 1/sqrt(100)
  const int Mrows = n * 25;
  const int ntiles = Mrows / 16;
  const int tail_rows = Mrows - ntiles * 16;
  const int tail_row0 = ntiles * 16;

  // y = 0
  fill_f32_kernel<<<(int)((nlf + 255) / 256), 256, 0, stream>>>(y, 0.0f, (long long)nlf);
  // edge preprocessing + y0 scatter
  edge_pre_kernel<<<(E + 255) / 256, 256, 0, stream>>>(idx, Z, disp, emb, W_rb, B0, SH, y, E);

  for (int s = 0; s < 2; ++s) {
    const float* Wq_s  = Wq  + s * 256;
    const float* Wk_s  = Wk  + s * 256;
    const float* Wv_s  = Wv  + s * 256;
    const float* Wo_s  = Wo  + s * 256;
    const float* Wb_s  = Wb  + s * 64;
    const float* Wg_s  = Wg  + s * 64;
    const float* Wg2_s = Wg2 + s * 64;

    // q/k/v projections via WMMA
    qkv_proj_wmma<<<512, 256, 0, stream>>>(y, Wq_s, Wk_s, Wv_s, Q, K, V, ntiles);
    if (tail_rows > 0) {
      int tb = (tail_rows * 16 + 255) / 256;
      proj_tail_kernel<<<tb, 256, 0, stream>>>(y, Wq_s, Q, tail_row0, tail_rows, 0);
      proj_tail_kernel<<<tb, 256, 0, stream>>>(y, Wk_s, K, tail_row0, tail_rows, 0);
      proj_tail_kernel<<<tb, 256, 0, stream>>>(y, Wv_s, V, tail_row0, tail_rows, 0);
    }

    // reset segment-softmax state and agg
    fill_f32_kernel<<<(int)((nlf + 255) / 256), 256, 0, stream>>>(AGG, 0.0f, (long long)nlf);
    fill_u32_kernel<<<(n * 4 + 255) / 256, 256, 0, stream>>>((unsigned int*)SMX, 0xFF800000u, (long long)n * 4);
    fill_f32_kernel<<<(n * 4 + 255) / 256, 256, 0, stream>>>(SSM, 0.0f, (long long)n * 4);

    attn_logits_kernel<<<(E + 255) / 256, 256, 0, stream>>>(idx, B0, Q, K, Wb_s, Wg_s, Wg2_s,
                                                            LG, G, G2, SMX, scale, E);
    attn_exp_kernel<<<(E * 4 + 255) / 256, 256, 0, stream>>>(idx, LG, SMX, SSM, E);

    long long total = (long long)E * 400;
    message_kernel<<<(int)((total + 255) / 256), 256, 0, stream>>>(idx, SH, G, G2, LG, SSM, V, AGG, total);

    // y += agg @ Wo via WMMA with C-accumulation
    wo_accum_wmma<<<512, 256, 0, stream>>>(AGG, Wo_s, y, ntiles);
    if (tail_rows > 0) {
      int tb = (tail_rows * 16 + 255) / 256;
      proj_tail_kernel<<<tb, 256, 0, stream>>>(AGG, Wo_s, y, tail_row0, tail_rows, 1);
    }
  }

  final_embed_kernel<<<(n * 16 + 255) / 256, 256, 0, stream>>>(Z, emb, W_embed, b_embed, y, n);
}